// GIN_64347200028751
// MI455X (gfx1250) — compile-verified
//
#include <hip/hip_runtime.h>
#include <hip/hip_bf16.h>
#include <cstddef>

// ---------------------------------------------------------------------------
// GIN on MI455X (gfx1250): fp32 WMMA (V_WMMA_F32_16X16X4_F32) for node GEMMs,
// hardware global_atomic_add_f32 for edge scatter-add and graph pooling.
// ---------------------------------------------------------------------------

#define N_NODES   100000
#define N_EDGES   1600000
#define DIM       128
#define OUT_CH    10
#define NUM_GRAPHS 128
#define BN_EPS    1e-5f
#define NB        ((size_t)N_NODES * DIM)       // 12,800,000 floats per node matrix
#define MTILES    (N_NODES / 16)                // 6250, exact

typedef float v2f __attribute__((ext_vector_type(2)));
typedef float v8f __attribute__((ext_vector_type(8)));

__device__ __forceinline__ void atomAddF(float* p, float v) {
    unsafeAtomicAdd(p, v);   // hardware global_atomic_add_f32 on CDNA
}

// ------------------------- simple utility kernels --------------------------

__global__ void k_zero(float* __restrict__ p, int n) {
    int i = blockIdx.x * blockDim.x + threadIdx.x;
    if (i < n) p[i] = 0.0f;
}

__global__ void k_copy4(float* __restrict__ dst, const float* __restrict__ src) {
    size_t i = (size_t)blockIdx.x * blockDim.x + threadIdx.x;   // one float4 each
    reinterpret_cast<float4*>(dst)[i] = reinterpret_cast<const float4*>(src)[i];
}

// ------------------------- edge scatter-add (GIN agg) ----------------------
// agg[dst] += x[src]; agg pre-initialized with x (eps=0 GIN: x + sum_neighbors)

__global__ void k_scatter(const float* __restrict__ x, const int* __restrict__ ei,
                          float* __restrict__ agg) {
    size_t idx = (size_t)blockIdx.x * blockDim.x + threadIdx.x;  // N_EDGES*32 threads
    int e     = (int)(idx >> 5);
    int chunk = (int)(idx & 31);
    int s = ei[e];
    int d = ei[N_EDGES + e];
    const float4 v = *reinterpret_cast<const float4*>(x + (size_t)s * DIM + chunk * 4);
    float* p = agg + (size_t)d * DIM + chunk * 4;
    atomAddF(p + 0, v.x);
    atomAddF(p + 1, v.y);
    atomAddF(p + 2, v.z);
    atomAddF(p + 3, v.w);
}

// ------------------------- fp32 WMMA GEMM ----------------------------------
// C[100000,128] = A[100000,128] @ W[128,128] + bias (+optional relu)
// One wave computes a 16x128 strip: 8 v8f accumulators, K-loop in steps of 4
// using V_WMMA_F32_16X16X4_F32.
//
// A fragment (16x4 f32, 2 VGPRs): lanes 0-15 -> M=lane, K pair {k0,k0+1};
//                                 lanes 16-31 -> M=lane-16, K pair {k0+2,k0+3}.
// B fragment (4x16 f32, 2 VGPRs): lanes 0-15 -> N=lane, K pair {k0,k0+1};
//                                 lanes 16-31 -> N=lane-16, K pair {k0+2,k0+3}.
// C/D (16x16 f32, 8 VGPRs): reg r -> M = r + 8*(lane>>4), N = lane&15.

__global__ void k_gemm128(const float* __restrict__ A, const float* __restrict__ W,
                          const float* __restrict__ bias, float* __restrict__ C,
                          int relu) {
    const int wave = threadIdx.x >> 5;          // 8 waves per block
    const int lane = threadIdx.x & 31;
    const int mtile = blockIdx.x * 8 + wave;
    if (mtile >= MTILES) return;                // wave-uniform: EXEC stays all-ones

    const int m0    = mtile * 16;
    const int half  = lane >> 4;                // 0 or 1 (K/M half selector)
    const int sub   = lane & 15;

    const float* Arow = A + (size_t)(m0 + sub) * DIM;

    v8f acc[8] = {};

    for (int k0 = 0; k0 < DIM; k0 += 4) {
        const int kk = k0 + 2 * half;
        v2f a = *reinterpret_cast<const v2f*>(Arow + kk);
        const float* Wrow0 = W + (size_t)kk * DIM;
        #pragma unroll
        for (int nt = 0; nt < 8; ++nt) {
            int n = nt * 16 + sub;
            v2f b;
            b.x = Wrow0[n];
            b.y = Wrow0[DIM + n];
            acc[nt] = __builtin_amdgcn_wmma_f32_16x16x4_f32(
                false, a, false, b, (short)0, acc[nt], false, false);
        }
    }

    #pragma unroll
    for (int nt = 0; nt < 8; ++nt) {
        int col = nt * 16 + sub;
        float bv = bias[col];
        #pragma unroll
        for (int r = 0; r < 8; ++r) {
            float v = acc[nt][r] + bv;
            if (relu) v = fmaxf(v, 0.0f);
            C[(size_t)(m0 + r + 8 * half) * DIM + col] = v;
        }
    }
}

// ------------------------- BatchNorm statistics ----------------------------
// Per-channel sum / sum-of-squares over the node axis (coalesced: thread=channel).

#define STAT_ROWS 500   // 200 blocks * 500 rows = 100000

__global__ void k_stats(const float* __restrict__ H, float* __restrict__ sums,
                        float* __restrict__ sumsq) {
    int c  = threadIdx.x;                // 0..127
    int r0 = blockIdx.x * STAT_ROWS;
    float s = 0.0f, q = 0.0f;
    for (int r = r0; r < r0 + STAT_ROWS; ++r) {
        float v = H[(size_t)r * DIM + c];
        s += v;
        q += v * v;
    }
    atomAddF(&sums[c], s);
    atomAddF(&sumsq[c], q);
}

// ------------------------- BN (train-mode) + ReLU --------------------------

__global__ void k_bnrelu(const float* __restrict__ H, const float* __restrict__ sums,
                         const float* __restrict__ sumsq, const float* __restrict__ gamma,
                         const float* __restrict__ beta, float* __restrict__ O) {
    size_t idx = (size_t)blockIdx.x * blockDim.x + threadIdx.x;  // one float4 each
    size_t off = idx * 4;
    int c = (int)(off & (DIM - 1));
    const float invN = 1.0f / (float)N_NODES;
    float4 h = *reinterpret_cast<const float4*>(H + off);
    float hv[4] = {h.x, h.y, h.z, h.w};
    float ov[4];
    #pragma unroll
    for (int j = 0; j < 4; ++j) {
        float mean = sums[c + j] * invN;
        float var  = sumsq[c + j] * invN - mean * mean;     // biased, matches jnp.var
        float sc   = gamma[c + j] * rsqrtf(var + BN_EPS);
        float v    = (hv[j] - mean) * sc + beta[c + j];
        ov[j] = fmaxf(v, 0.0f);
    }
    float4 o = {ov[0], ov[1], ov[2], ov[3]};
    *reinterpret_cast<float4*>(O + off) = o;
}

// ------------------------- graph pooling -----------------------------------

__global__ void k_pool(const float* __restrict__ H, const int* __restrict__ batch,
                       float* __restrict__ pooled) {
    size_t idx = (size_t)blockIdx.x * blockDim.x + threadIdx.x;  // N_NODES*32 threads
    int node  = (int)(idx >> 5);
    int chunk = (int)(idx & 31);
    int g = batch[node];
    const float4 v = *reinterpret_cast<const float4*>(H + (size_t)node * DIM + chunk * 4);
    float* p = pooled + (size_t)g * DIM + chunk * 4;
    atomAddF(p + 0, v.x);
    atomAddF(p + 1, v.y);
    atomAddF(p + 2, v.z);
    atomAddF(p + 3, v.w);
}

// ------------------------- tiny readout FCs --------------------------------

__global__ void k_fc(const float* __restrict__ X, const float* __restrict__ Wm,
                     const float* __restrict__ b, float* __restrict__ Y,
                     int M, int K, int N, int relu) {
    int idx = blockIdx.x * blockDim.x + threadIdx.x;
    if (idx >= M * N) return;
    int m = idx / N, n = idx % N;
    float s = b[n];
    for (int k = 0; k < K; ++k) s += X[m * K + k] * Wm[k * N + n];
    Y[idx] = relu ? fmaxf(s, 0.0f) : s;
}

// ---------------------------------------------------------------------------

extern "C" void kernel_launch(void* const* d_in, const int* in_sizes, int n_in,
                              void* d_out, int out_size, void* d_ws, size_t ws_size,
                              hipStream_t stream) {
    const float* x    = (const float*)d_in[0];
    const int*   ei   = (const int*)d_in[1];
    const int*   batch= (const int*)d_in[2];
    const float* W1a  = (const float*)d_in[3];
    const float* b1a  = (const float*)d_in[4];
    const float* ga   = (const float*)d_in[5];
    const float* ba   = (const float*)d_in[6];
    const float* W2a  = (const float*)d_in[7];
    const float* b2a  = (const float*)d_in[8];
    const float* W1b  = (const float*)d_in[9];
    const float* b1b  = (const float*)d_in[10];
    const float* gb   = (const float*)d_in[11];
    const float* bb   = (const float*)d_in[12];
    const float* W2b  = (const float*)d_in[13];
    const float* b2b  = (const float*)d_in[14];
    const float* Wl1  = (const float*)d_in[15];
    const float* bl1  = (const float*)d_in[16];
    const float* Wl2  = (const float*)d_in[17];
    const float* bl2  = (const float*)d_in[18];

    float* buf0   = (float*)d_ws;
    float* buf1   = buf0 + NB;
    float* sums   = buf1 + NB;
    float* sumsq  = sums + DIM;
    float* pooled = sumsq + DIM;
    float* tmp    = pooled + NUM_GRAPHS * DIM;

    const int TB = 256;
    const int copyBlocks    = (int)(NB / 4 / TB);            // 12500
    const int scatterBlocks = (int)((size_t)N_EDGES * 32 / TB); // 200000
    const int gemmBlocks    = (MTILES + 7) / 8;              // 782
    const int bnBlocks      = (int)(NB / 4 / TB);            // 12500
    const int poolBlocks    = (int)((size_t)N_NODES * 32 / TB); // 12500

    // ---------------- conv1 ----------------
    k_copy4  <<<copyBlocks, TB, 0, stream>>>(buf1, x);                  // agg := x
    k_scatter<<<scatterBlocks, TB, 0, stream>>>(x, ei, buf1);           // agg += x[src]->dst
    k_gemm128<<<gemmBlocks, TB, 0, stream>>>(buf1, W1a, b1a, buf0, 0);  // h1
    k_zero   <<<1, DIM, 0, stream>>>(sums, DIM);
    k_zero   <<<1, DIM, 0, stream>>>(sumsq, DIM);
    k_stats  <<<N_NODES / STAT_ROWS, DIM, 0, stream>>>(buf0, sums, sumsq);
    k_bnrelu <<<bnBlocks, TB, 0, stream>>>(buf0, sums, sumsq, ga, ba, buf1);
    k_gemm128<<<gemmBlocks, TB, 0, stream>>>(buf1, W2a, b2a, buf0, 1);  // h2

    // ---------------- conv2 ----------------
    k_copy4  <<<copyBlocks, TB, 0, stream>>>(buf1, buf0);               // agg := h2
    k_scatter<<<scatterBlocks, TB, 0, stream>>>(buf0, ei, buf1);
    k_gemm128<<<gemmBlocks, TB, 0, stream>>>(buf1, W1b, b1b, buf0, 0);  // h3
    k_zero   <<<1, DIM, 0, stream>>>(sums, DIM);
    k_zero   <<<1, DIM, 0, stream>>>(sumsq, DIM);
    k_stats  <<<N_NODES / STAT_ROWS, DIM, 0, stream>>>(buf0, sums, sumsq);
    k_bnrelu <<<bnBlocks, TB, 0, stream>>>(buf0, sums, sumsq, gb, bb, buf1);
    k_gemm128<<<gemmBlocks, TB, 0, stream>>>(buf1, W2b, b2b, buf0, 1);  // h4

    // ---------------- pool + readout ----------------
    k_zero<<<(NUM_GRAPHS * DIM + TB - 1) / TB, TB, 0, stream>>>(pooled, NUM_GRAPHS * DIM);
    k_pool<<<poolBlocks, TB, 0, stream>>>(buf0, batch, pooled);
    k_fc  <<<(NUM_GRAPHS * DIM + TB - 1) / TB, TB, 0, stream>>>(pooled, Wl1, bl1, tmp,
                                                                NUM_GRAPHS, DIM, DIM, 1);
    k_fc  <<<(NUM_GRAPHS * OUT_CH + TB - 1) / TB, TB, 0, stream>>>(tmp, Wl2, bl2,
                                                                   (float*)d_out,
                                                                   NUM_GRAPHS, DIM, OUT_CH, 0);
}